// BaseFasterRCNN_12979391168525
// MI455X (gfx1250) — compile-verified
//
#include <hip/hip_runtime.h>
#include <math.h>
#include <stdint.h>

// ---------------- problem constants ----------------
#define R            2000
#define NCLS         21
#define NC           20          // foreground classes
#define RP           2016        // R padded to multiple of 32 (126 tiles of 16)
#define NTILE        126         // RP / 16
#define WSTRIDE      64          // u32 words per bitmask row (63 used, padded)
#define SCORE_THRESH 0.05f
#define NMS_THRESH   0.3f

typedef float v2f __attribute__((ext_vector_type(2)));
typedef float v8f __attribute__((ext_vector_type(8)));

// ---------------- K1: softmax over 21 classes per ROI ----------------
__global__ void softmax_kernel(const float* __restrict__ scores,
                               float* __restrict__ prob) {
    int r = blockIdx.x * blockDim.x + threadIdx.x;
    if (r >= R) return;
    const float* s = scores + r * NCLS;
    float m = s[0];
#pragma unroll
    for (int c = 1; c < NCLS; ++c) m = fmaxf(m, s[c]);
    float e[NCLS];
    float sum = 0.f;
#pragma unroll
    for (int c = 0; c < NCLS; ++c) { e[c] = __expf(s[c] - m); sum += e[c]; }
    float inv = 1.0f / sum;
    float* p = prob + r * NCLS;
#pragma unroll
    for (int c = 0; c < NCLS; ++c) p[c] = e[c] * inv;
}

// ---------------- K2: per-class rank (argsort by -score, stable) + decode +
//                  scatter sorted boxes / areas / probs / order -------------
__global__ void rank_decode_kernel(const float* __restrict__ loc_in,
                                   const float* __restrict__ rois,
                                   const float* __restrict__ prob,
                                   const int* __restrict__ sh_p,
                                   const int* __restrict__ sw_p,
                                   float4* __restrict__ sbox,   // [NC][RP]
                                   float*  __restrict__ sarea,  // [NC][RP]
                                   float*  __restrict__ sprob,  // [NC][R]
                                   int*    __restrict__ order)  // [NC][R]
{
    __shared__ float sc[R];
    int cls = blockIdx.x;        // 0..19 -> class c = cls+1
    int c   = cls + 1;
    int i   = blockIdx.y * blockDim.x + threadIdx.x;

    for (int t = threadIdx.x; t < R; t += blockDim.x)
        sc[t] = prob[t * NCLS + c];
    __syncthreads();

    if (i >= R) {                 // zero the 16-entry pad so padded tiles are inert
        if (i < RP) { sbox[cls * RP + i] = make_float4(0.f, 0.f, 0.f, 0.f);
                      sarea[cls * RP + i] = 0.f; }
        return;
    }

    // rank = #{j : s_j > s_i  or (s_j == s_i and j < i)}  (stable descending)
    float si = sc[i];
    int rank = 0;
    for (int j = 0; j < R; ++j) {
        float sj = sc[j];
        rank += (sj > si) || (sj == si && j < i);
    }

    // decode + clip this (roi, class) box
    float size_h = (float)*sh_p, size_w = (float)*sw_p;
    float4 roi = ((const float4*)rois)[i];
    float h  = roi.z - roi.x, w = roi.w - roi.y;
    float cy = roi.x + 0.5f * h, cx = roi.y + 0.5f * w;
    const float* lp = loc_in + (size_t)i * (NCLS * 4) + c * 4;
    float dy = lp[0] * 0.1f, dx = lp[1] * 0.1f;
    float dh = lp[2] * 0.2f, dw = lp[3] * 0.2f;
    float ncy = dy * h + cy, ncx = dx * w + cx;
    float nh = __expf(dh) * h, nw = __expf(dw) * w;
    float y0 = fminf(fmaxf(ncy - 0.5f * nh, 0.f), size_h);
    float x0 = fminf(fmaxf(ncx - 0.5f * nw, 0.f), size_w);
    float y1 = fminf(fmaxf(ncy + 0.5f * nh, 0.f), size_h);
    float x1 = fminf(fmaxf(ncx + 0.5f * nw, 0.f), size_w);

    int dst = cls * RP + rank;
    sbox[dst]  = make_float4(y0, x0, y1, x1);
    sarea[dst] = (y1 - y0) * (x1 - x0);
    sprob[cls * R + rank] = si;
    order[cls * R + rank] = i;
}

// uniform 8-way select (values are wave-uniform ballots; avoids scratch)
__device__ __forceinline__ unsigned sel8(const unsigned b[8], int k) {
    unsigned s0 = (k & 1) ? b[1] : b[0];
    unsigned s1 = (k & 1) ? b[3] : b[2];
    unsigned s2 = (k & 1) ? b[5] : b[4];
    unsigned s3 = (k & 1) ? b[7] : b[6];
    unsigned t0 = (k & 2) ? s1 : s0;
    unsigned t1 = (k & 2) ? s3 : s2;
    return (k & 4) ? t1 : t0;
}

// One 16x16 tile: WMMA produces area_i only in C/D layout; VALU computes
// intersections; returns 8 ballots (VGPR v packs rows v / v+8 in lo/hi 16).
__device__ __forceinline__ void tile_ballots(const float4* __restrict__ lb,
                                             const float*  __restrict__ la,
                                             int colBase, bool hi, int ln,
                                             const float4 bi[8], const int ig[8],
                                             v2f A, unsigned bal[8])
{
    int j = colBase + ln;
    float4 bj = lb[j];
    // B (4x16 f32): row K=0 -> ones, row K=2 -> area_j, rows K=1/3 -> 0
    v2f B; B.x = hi ? la[j] : 1.0f;
           B.y = 0.f;
    v8f Z = {};
    v8f d = __builtin_amdgcn_wmma_f32_16x16x4_f32(
        false, A, false, B, (short)0, Z, false, false);
#pragma unroll
    for (int v = 0; v < 8; ++v) {
        float ty1 = fmaxf(bi[v].x, bj.x);
        float tx1 = fmaxf(bi[v].y, bj.y);
        float ty2 = fminf(bi[v].z, bj.z);
        float tx2 = fminf(bi[v].w, bj.w);
        float ih = fmaxf(ty2 - ty1, 0.f);
        float iw = fmaxf(tx2 - tx1, 0.f);
        float inter = ih * iw;
        float uni = fmaxf(d[v] - inter, 1e-10f);
        bool pred = (j < ig[v]) && (inter > NMS_THRESH * uni);
        bal[v] = (unsigned)__ballot(pred);
    }
}

// ---------------- K3: suppression bitmask via WMMA tiles -------------------
// LDS staging uses CDNA5 async loads (GLOBAL_LOAD_ASYNC_TO_LDS_*, ASYNCcnt):
// global data flows straight into LDS without a VGPR round-trip. Each wave
// then owns a 16-row stripe (sorted order) and sweeps 63 column-word steps,
// two 16-col WMMA tiles per step.
__global__ __launch_bounds__(256)
void iou_bitmask_kernel(const float4* __restrict__ sbox,
                        const float*  __restrict__ sarea,
                        unsigned* __restrict__ sup)          // [NC][R][WSTRIDE]
{
    __shared__ float4 lb[RP];
    __shared__ float  la[RP];
    int cls = blockIdx.x;

    // async DMA: global -> LDS, per-lane; vdst = LDS byte offset (low 32 bits
    // of the flat shared address per the ISA aperture rule), GV addressing.
    for (int t = threadIdx.x; t < RP; t += blockDim.x) {
        unsigned llb = (unsigned)(size_t)&lb[t];
        const float4* gb = sbox + (size_t)cls * RP + t;
        asm volatile("global_load_async_to_lds_b128 %0, %1, off"
                     :: "v"(llb), "v"(gb) : "memory");
        unsigned lla = (unsigned)(size_t)&la[t];
        const float* ga = sarea + (size_t)cls * RP + t;
        asm volatile("global_load_async_to_lds_b32 %0, %1, off"
                     :: "v"(lla), "v"(ga) : "memory");
    }
    asm volatile("s_wait_asynccnt 0" ::: "memory");
    __syncthreads();

    int wave = threadIdx.x >> 5;
    int lane = threadIdx.x & 31;
    int rowTile = blockIdx.y * 8 + wave;
    if (rowTile >= (R / 16)) return;      // wave-uniform exit; 125 live stripes
    int iBase = rowTile * 16;
    bool hi = lane >= 16;
    int  ln = lane & 15;

    // this lane's 8 row boxes (M = v + 8*hi per WMMA C/D layout)
    float4 bi[8]; int ig[8];
#pragma unroll
    for (int v = 0; v < 8; ++v) {
        int idx = iBase + v + (hi ? 8 : 0);
        bi[v] = lb[idx];
        ig[v] = idx;
    }

    // A (16x4 f32): row M -> [area_M, 1, 0, 0]; lanes 0-15 hold K={0,1}
    v2f A; A.x = hi ? 0.f : la[iBase + ln];
           A.y = hi ? 0.f : 1.0f;

    for (int tw = 0; tw < NTILE / 2; ++tw) {     // one u32 word per step
        unsigned balE[8], balO[8];
        tile_ballots(lb, la, tw * 32,      hi, ln, bi, ig, A, balE);
        tile_ballots(lb, la, tw * 32 + 16, hi, ln, bi, ig, A, balO);
        int k = lane & 7;
        unsigned be = sel8(balE, k);
        unsigned bo = sel8(balO, k);
        unsigned word = (lane < 8) ? ((be & 0xFFFFu) | (bo << 16))
                                   : ((be >> 16) | (bo & 0xFFFF0000u));
        if (lane < 16)
            sup[((size_t)cls * R + iBase + lane) * WSTRIDE + tw] = word;
    }
}

// ---------------- K4: sequential greedy scan (1 wave / class) --------------
__global__ void nms_scan_kernel(const unsigned* __restrict__ sup,
                                const float* __restrict__ sprob,
                                unsigned* __restrict__ kept)   // [NC][WSTRIDE]
{
    int cls  = blockIdx.x;
    int lane = threadIdx.x;                   // 32 threads
    unsigned k0 = 0, k1 = 0;                  // kept bitset: words lane, lane+32
    const unsigned* base = sup + (size_t)cls * R * WSTRIDE;
    for (int i = 0; i < R; ++i) {
        const unsigned* row = base + (size_t)i * WSTRIDE;
        __builtin_prefetch(row + WSTRIDE, 0, 0);           // global_prefetch_b8
        __builtin_prefetch(row + 2 * WSTRIDE, 0, 0);
        unsigned w0 = row[lane];
        unsigned w1 = row[lane + 32];
        unsigned hitw = (w0 & k0) | (w1 & k1);
        bool any = (__ballot(hitw != 0u) != 0ull);
        bool keep = (sprob[cls * R + i] > SCORE_THRESH) && !any;
        if (keep) {
            int wi = i >> 5;
            unsigned bit = 1u << (i & 31);
            if (wi < 32) { if (lane == wi)        k0 |= bit; }
            else         { if (lane == (wi - 32)) k1 |= bit; }
        }
    }
    kept[cls * WSTRIDE + lane]      = k0;
    kept[cls * WSTRIDE + lane + 32] = k1;
}

// ---------------- K5: scatter to output (20, 2000, 5) ----------------------
__global__ void output_kernel(const float4* __restrict__ sbox,
                              const float* __restrict__ sprob,
                              const int* __restrict__ order,
                              const unsigned* __restrict__ kept,
                              float* __restrict__ out)
{
    int cls = blockIdx.x;
    int s = blockIdx.y * blockDim.x + threadIdx.x;
    if (s >= R) return;
    int orig = order[cls * R + s];
    unsigned kw = kept[cls * WSTRIDE + (s >> 5)];
    float k = ((kw >> (s & 31)) & 1u) ? 1.0f : 0.0f;
    float4 b = sbox[cls * RP + s];
    float p = sprob[cls * R + s];
    float* o = out + ((size_t)cls * R + orig) * 5;
    o[0] = b.x * k; o[1] = b.y * k; o[2] = b.z * k; o[3] = b.w * k; o[4] = p * k;
}

// ---------------- launcher -------------------------------------------------
extern "C" void kernel_launch(void* const* d_in, const int* in_sizes, int n_in,
                              void* d_out, int out_size, void* d_ws, size_t ws_size,
                              hipStream_t stream) {
    const float* roi_cls_loc = (const float*)d_in[0];
    const float* roi_scores  = (const float*)d_in[1];
    const float* rois        = (const float*)d_in[2];
    const int*   sh_p        = (const int*)d_in[3];
    const int*   sw_p        = (const int*)d_in[4];

    char* p = (char*)d_ws;
    float*    prob  = (float*)p;    p += sizeof(float) * R * NCLS;        // 168000 B
    float4*   sbox  = (float4*)p;   p += sizeof(float4) * NC * RP;        // 645120 B
    float*    sarea = (float*)p;    p += sizeof(float) * NC * RP;         // 161280 B
    float*    sprob = (float*)p;    p += sizeof(float) * NC * R;          // 160000 B
    int*      order = (int*)p;      p += sizeof(int)   * NC * R;          // 160000 B
    unsigned* kept  = (unsigned*)p; p += sizeof(unsigned) * NC * WSTRIDE; //   5120 B
    unsigned* sup   = (unsigned*)p;                       // 20*2000*64*4 = 10.24 MB

    softmax_kernel<<<dim3((R + 255) / 256), 256, 0, stream>>>(roi_scores, prob);

    rank_decode_kernel<<<dim3(NC, (R + 255) / 256), 256, 0, stream>>>(
        roi_cls_loc, rois, prob, sh_p, sw_p, sbox, sarea, sprob, order);

    iou_bitmask_kernel<<<dim3(NC, ((R / 16) + 7) / 8), 256, 0, stream>>>(
        sbox, sarea, sup);

    nms_scan_kernel<<<dim3(NC), 32, 0, stream>>>(sup, sprob, kept);

    output_kernel<<<dim3(NC, (R + 255) / 256), 256, 0, stream>>>(
        sbox, sprob, order, kept, (float*)d_out);
}